// Model_79568564126463
// MI455X (gfx1250) — compile-verified
//
#include <hip/hip_runtime.h>
#include <hip/hip_bf16.h>

#define T_N     4000000
#define DT_F    0.0001f
#define NCHUNK  32768      // chunks (power of 2)
#define LCH     126        // steps per chunk (multiple of 7); 32768*126 >= T_N
#define NGRP    (LCH / 7)  // 18 WMMA groups per chunk
#define SC      (NCHUNK / 1024)  // chunks per thread in pass2

typedef float v2f __attribute__((ext_vector_type(2)));
typedef float v8f __attribute__((ext_vector_type(8)));

// ---------- 2x2 matrix helpers (row major: [a b; c d]) ----------
struct M2 { float a, b, c, d; };

__device__ __forceinline__ M2 mmul(M2 x, M2 y) {
    return { x.a*y.a + x.b*y.c, x.a*y.b + x.b*y.d,
             x.c*y.a + x.d*y.c, x.c*y.b + x.d*y.d };
}
__device__ __forceinline__ M2 mpow(M2 m, unsigned e) {
    M2 r = {1.f, 0.f, 0.f, 1.f};
    while (e) { if (e & 1u) r = mmul(r, m); m = mmul(m, m); e >>= 1; }
    return r;
}

struct Params { M2 Trans, XiCov, Cout; float p0, p1; };

__device__ __forceinline__ Params derive(const float* A, const float* B,
                                         const float* E, const float* cov,
                                         const float* tp) {
    const float s2 = 1.41421356237309515f;
    M2 a  = {A[0], A[1], A[2], A[3]};
    M2 b  = {B[0], B[1], B[2], B[3]};
    M2 e  = {E[0], E[1], E[2], E[3]};
    M2 ci = {cov[0], cov[1], cov[2], cov[3]};
    M2 cb = mmul(ci, b);
    M2 Xi = {(e.a - cb.a)/s2, (e.b - cb.b)/s2, (e.c - cb.c)/s2, (e.d - cb.d)/s2};
    M2 nbt = {-s2*b.a, -s2*b.c, -s2*b.b, -s2*b.d};  // -sqrt2 * B^T
    M2 XiC = mmul(Xi, nbt);
    M2 Tr = {1.f + (a.a - XiC.a)*DT_F, (a.b - XiC.b)*DT_F,
             (a.c - XiC.c)*DT_F, 1.f + (a.d - XiC.d)*DT_F};
    M2 Co = {nbt.a*DT_F, nbt.b*DT_F, nbt.c*DT_F, nbt.d*DT_F};
    Params p; p.Trans = Tr; p.XiCov = Xi; p.Cout = Co; p.p0 = tp[0]; p.p1 = tp[1];
    return p;
}

// ---------- WMMA: D(16x16,f32) = A(16x4,f32) x B(4x16,f32) + C ----------
__device__ __forceinline__ v8f wmma4(v2f a, v2f b, v8f c) {
    return __builtin_amdgcn_wmma_f32_16x16x4_f32(false, a, false, b,
                                                 (short)0, c, false, false);
}

// ================= Pass 1: per-chunk summaries (scan from x=0) =================
__global__ __launch_bounds__(256)
void lscan_pass1(const float* __restrict__ in, const float* __restrict__ A,
                 const float* __restrict__ B, const float* __restrict__ E,
                 const float* __restrict__ cov, const float* __restrict__ tp,
                 float* __restrict__ csum) {
    int chunk = blockIdx.x * blockDim.x + threadIdx.x;
    if (chunk >= NCHUNK) return;
    Params P = derive(A, B, E, cov, tp);
    float x0 = 0.f, x1 = 0.f;
    int base = chunk * LCH;
    for (int j = 0; j < LCH; ++j) {
        int tt = base + j;
        if (tt >= T_N) break;
        // stream prefetch ~2 cachelines ahead (L2 is cold on first pass)
        int pf = tt + 24; pf = pf < T_N ? pf : (T_N - 1);
        __builtin_prefetch(in + pf * 3, 0, 3);
        const float* r = in + tt * 3;
        float t = r[0], dy0 = r[1], dy1 = r[2];
        float s = P.p0 * __cosf(P.p1 * t) * DT_F;
        float n0 = P.Trans.a*x0 + P.Trans.b*x1 + P.XiCov.a*dy0 + P.XiCov.b*dy1 + s;
        float n1 = P.Trans.c*x0 + P.Trans.d*x1 + P.XiCov.c*dy0 + P.XiCov.d*dy1;
        x0 = n0; x1 = n1;
    }
    csum[2*chunk + 0] = x0;
    csum[2*chunk + 1] = x1;
}

// ======== Pass 2: prefix scan over chunk summaries + build W matrix ========
__global__ __launch_bounds__(1024)
void lscan_pass2(const float* __restrict__ A, const float* __restrict__ B,
                 const float* __restrict__ E, const float* __restrict__ cov,
                 const float* __restrict__ tp, const float* __restrict__ init,
                 const float* __restrict__ csum, float* __restrict__ xs,
                 float* __restrict__ Wm) {
    __shared__ float sx0[1024];
    __shared__ float sx1[1024];
    int i = threadIdx.x;
    Params P = derive(A, B, E, cov, tp);
    M2 ML  = mpow(P.Trans, LCH);   // per-chunk transition
    M2 M32 = mpow(ML, SC);         // per-thread-segment transition

    // serial fold over my SC chunks
    float x0 = 0.f, x1 = 0.f;
    if (i == 0) { x0 = init[0]; x1 = init[1]; }
    for (int k = 0; k < SC; ++k) {
        int ch = i * SC + k;
        float c0 = csum[2*ch], c1 = csum[2*ch + 1];
        float n0 = ML.a*x0 + ML.b*x1 + c0;
        float n1 = ML.c*x0 + ML.d*x1 + c1;
        x0 = n0; x1 = n1;
    }
    sx0[i] = x0; sx1[i] = x1;
    __syncthreads();

    // Hillis-Steele inclusive scan; combine across 2^d uses M32^(2^d)
    M2 Pd = M32;
    for (int d = 0; d < 10; ++d) {
        int ofs = 1 << d;
        float a0 = 0.f, a1 = 0.f;
        if (i >= ofs) { a0 = sx0[i - ofs]; a1 = sx1[i - ofs]; }
        __syncthreads();
        if (i >= ofs) {
            x0 = Pd.a*a0 + Pd.b*a1 + x0;
            x1 = Pd.c*a0 + Pd.d*a1 + x1;
        }
        sx0[i] = x0; sx1[i] = x1;
        __syncthreads();
        Pd = mmul(Pd, Pd);
    }

    // exclusive prefix for my segment
    float e0, e1;
    if (i == 0) { e0 = init[0]; e1 = init[1]; }
    else        { e0 = sx0[i-1]; e1 = sx1[i-1]; }

    // replay: emit per-chunk start states
    for (int k = 0; k < SC; ++k) {
        int ch = i * SC + k;
        xs[2*ch + 0] = e0; xs[2*ch + 1] = e1;
        float c0 = csum[2*ch], c1 = csum[2*ch + 1];
        float n0 = ML.a*e0 + ML.b*e1 + c0;
        float n1 = ML.c*e0 + ML.d*e1 + c1;
        e0 = n0; e1 = n1;
    }

    // thread 0 builds the 16x16 group matrix W:
    //   v = [x; b0..b6] -> [out0..out6 ; x_next]
    if (i == 0) {
        M2 Pj[8];
        Pj[0] = {1.f, 0.f, 0.f, 1.f};
        for (int j = 1; j < 8; ++j) Pj[j] = mmul(P.Trans, Pj[j-1]);
        M2 CT[7];
        for (int j = 0; j < 7; ++j) CT[j] = mmul(P.Cout, Pj[j]);
        for (int k = 0; k < 256; ++k) Wm[k] = 0.f;
        auto put = [&](M2 M, int r, int c) {
            Wm[r*16 + c]       = M.a; Wm[r*16 + c + 1]       = M.b;
            Wm[(r+1)*16 + c]   = M.c; Wm[(r+1)*16 + c + 1]   = M.d;
        };
        for (int j = 0; j < 7; ++j) {
            int r = 2*j;
            put(CT[j], r, 0);                               // x columns
            for (int q = 0; q < j; ++q)
                put(CT[j-1-q], r, 2 + 2*q);                 // b_q columns
        }
        put(Pj[7], 14, 0);
        for (int q = 0; q < 7; ++q)
            put(Pj[6-q], 14, 2 + 2*q);
    }
}

// ============ Pass 3: WMMA-powered output pass (16 chunks / wave) ============
__global__ __launch_bounds__(256)
void lscan_pass3(const float* __restrict__ in, const float* __restrict__ A,
                 const float* __restrict__ B, const float* __restrict__ E,
                 const float* __restrict__ cov, const float* __restrict__ tp,
                 const float* __restrict__ xs, const float* __restrict__ Wm,
                 float* __restrict__ out) {
    int tid  = blockIdx.x * blockDim.x + threadIdx.x;
    int wave = tid >> 5;
    int lane = tid & 31;
    int m    = lane & 15;   // chunk slot / matrix row
    int hi   = lane >> 4;   // half of the wave

    Params P = derive(A, B, E, cov, tp);
    float p0 = P.p0, p1 = P.p1;
    M2 Xi = P.XiCov;

    // A-operand: W columns, 4 K-chunks. lanes0-15 hold K=0,1; lanes16-31 K=2,3.
    v2f a0 = { Wm[m*16 + 0  + 2*hi], Wm[m*16 + 1  + 2*hi] };
    v2f a1 = { Wm[m*16 + 4  + 2*hi], Wm[m*16 + 5  + 2*hi] };
    v2f a2 = { Wm[m*16 + 8  + 2*hi], Wm[m*16 + 9  + 2*hi] };
    v2f a3 = { Wm[m*16 + 12 + 2*hi], Wm[m*16 + 13 + 2*hi] };

    int chunk = wave * 16 + m;
    int base  = chunk * LCH;

    float x0 = 0.f, x1 = 0.f;
    if (!hi) { x0 = xs[2*chunk]; x1 = xs[2*chunk + 1]; }

    v2f* o2 = (v2f*)out;

    auto makeb = [&](int tt, float& b0o, float& b1o) {
        int rr = tt < T_N ? tt : (T_N - 1);     // clamp (out-of-range never stored)
        const float* r = in + rr * 3;
        float t = r[0], dy0 = r[1], dy1 = r[2];
        float s = p0 * __cosf(p1 * t) * DT_F;
        b0o = Xi.a*dy0 + Xi.b*dy1 + s;
        b1o = Xi.c*dy0 + Xi.d*dy1;
    };

    // B-operand b-values: lower half supplies rows {x, b1, b3, b5},
    // upper half supplies rows {b0, b2, b4, b6}. Double-buffered so next
    // group's loads overlap the current group's WMMA chain.
    auto computeB = [&](int tb, float* e) {
        if (hi) {
            makeb(tb + 0, e[0], e[1]);
            makeb(tb + 2, e[2], e[3]);
            makeb(tb + 4, e[4], e[5]);
            makeb(tb + 6, e[6], e[7]);
        } else {
            e[0] = 0.f; e[1] = 0.f;             // slot 0 = x, inserted at use
            makeb(tb + 1, e[2], e[3]);
            makeb(tb + 3, e[4], e[5]);
            makeb(tb + 5, e[6], e[7]);
        }
    };

    float cur[8];
    computeB(base, cur);

    for (int g = 0; g < NGRP; ++g) {
        int tb = base + 7 * g;   // timestep of out0 in this group

        v2f B0 = hi ? v2f{cur[0], cur[1]} : v2f{x0, x1};
        v2f B1 = {cur[2], cur[3]};
        v2f B2 = {cur[4], cur[5]};
        v2f B3 = {cur[6], cur[7]};

        v8f acc = {0.f, 0.f, 0.f, 0.f, 0.f, 0.f, 0.f, 0.f};
        acc = wmma4(a0, B0, acc);
        acc = wmma4(a1, B1, acc);
        acc = wmma4(a2, B2, acc);
        acc = wmma4(a3, B3, acc);

        // prefetch 2 groups ahead, then issue next group's loads now
        // (independent of the WMMA result -> overlaps the matrix pipe)
        {
            int pf = tb + 14; pf = pf < T_N ? pf : (T_N - 1);
            __builtin_prefetch(in + pf * 3, 0, 3);
        }
        float nxt[8];
        computeB(tb + 7, nxt);

        // D layout: lanes0-15 hold rows 0-7 = out[0..3]; lanes16-31 hold
        // rows 8-15 = out[4..6] and x_next (VGPRs 6,7).
        // Output is write-once streaming -> non-temporal stores keep the
        // 48MB input resident in the 192MB L2 between passes.
        if (!hi) {
            if (tb + 3 < T_N) {
                __builtin_nontemporal_store(v2f{acc[0], acc[1]}, o2 + tb + 0);
                __builtin_nontemporal_store(v2f{acc[2], acc[3]}, o2 + tb + 1);
                __builtin_nontemporal_store(v2f{acc[4], acc[5]}, o2 + tb + 2);
                __builtin_nontemporal_store(v2f{acc[6], acc[7]}, o2 + tb + 3);
            } else {
                for (int j = 0; j < 4; ++j) {
                    if (tb + j < T_N)
                        __builtin_nontemporal_store(v2f{acc[2*j], acc[2*j+1]},
                                                    o2 + tb + j);
                }
            }
        } else {
            if (tb + 6 < T_N) {
                __builtin_nontemporal_store(v2f{acc[0], acc[1]}, o2 + tb + 4);
                __builtin_nontemporal_store(v2f{acc[2], acc[3]}, o2 + tb + 5);
                __builtin_nontemporal_store(v2f{acc[4], acc[5]}, o2 + tb + 6);
            } else {
                for (int j = 4; j < 7; ++j) {
                    if (tb + j < T_N)
                        __builtin_nontemporal_store(
                            v2f{acc[2*(j-4)], acc[2*(j-4)+1]}, o2 + tb + j);
                }
            }
        }

        // ship x_next from upper-half lane back to lower-half lane
        float nx0 = __shfl(acc[6], lane | 16, 32);
        float nx1 = __shfl(acc[7], lane | 16, 32);
        x0 = nx0; x1 = nx1;

        #pragma unroll
        for (int q = 0; q < 8; ++q) cur[q] = nxt[q];
    }
}

// =============================== launcher ===============================
extern "C" void kernel_launch(void* const* d_in, const int* in_sizes, int n_in,
                              void* d_out, int out_size, void* d_ws, size_t ws_size,
                              hipStream_t stream) {
    const float* in   = (const float*)d_in[0];
    const float* A    = (const float*)d_in[1];
    const float* B    = (const float*)d_in[2];
    // d_in[3] = D (unused by reference)
    const float* E    = (const float*)d_in[4];
    const float* cov  = (const float*)d_in[5];
    const float* tp   = (const float*)d_in[6];
    const float* init = (const float*)d_in[7];
    float* out = (float*)d_out;

    float* ws   = (float*)d_ws;
    float* csum = ws;                 // [2*NCHUNK]
    float* xs   = ws + 2 * NCHUNK;    // [2*NCHUNK]
    float* Wm   = ws + 4 * NCHUNK;    // [256]

    lscan_pass1<<<NCHUNK / 256, 256, 0, stream>>>(in, A, B, E, cov, tp, csum);
    lscan_pass2<<<1, 1024, 0, stream>>>(A, B, E, cov, tp, init, csum, xs, Wm);
    lscan_pass3<<<(NCHUNK / 16) * 32 / 256, 256, 0, stream>>>(in, A, B, E, cov, tp,
                                                              xs, Wm, out);
}